// ContradictionResolver_16527034155597
// MI455X (gfx1250) — compile-verified
//
#include <hip/hip_runtime.h>
#include <hip/hip_bf16.h>

// ContradictionResolver for MI455X (gfx1250, wave32).
// B=4, N=512, F=128.
//
// Pipeline (all on `stream`):
//  k_proj : Hi = H*dW1[:F] + db1 (row-major), HjT = (H*dW1[F:])^T  [WMMA f32 16x16x4]
//  k_pair : pairwise ELU-dot scores, masked row-max -> maxc, degree -> deg,
//           writes bool mask (as 0/1 float) to the tail of d_out        [VALU/TRANS]
//  k_nbr  : nbrF = ((A>0)*H) / max(deg,1)                               [WMMA]
//  k_res1 : res1 = elu(H*rW1[0:128] + nbrF*rW1[128:256] + maxc*rW1[256] + rb1) [WMMA]
//  k_res2 : res2 = res1*rW2 + rb2 ; out = update ? res2 : H             [WMMA]

#define BB 4
#define NN 512
#define FF 128

typedef float v2f __attribute__((ext_vector_type(2)));
typedef float v8f __attribute__((ext_vector_type(8)));

// ---- WMMA f32 16x16x4 fragment helpers (ISA 7.12.2 layouts) ------------------
// A (16x4, MxK, row-major src): lanes 0-15 -> M=lane, VGPR0/1 = K0/K1;
//                               lanes 16-31 -> M=lane-16, VGPR0/1 = K2/K3.
__device__ __forceinline__ v2f ldA(const float* S, int ld, int r0, int k0, int lane) {
  int m  = lane & 15;
  int kk = k0 + ((lane >> 4) << 1);
  const float* p = S + (size_t)(r0 + m) * ld + kk;
  v2f a; a.x = p[0]; a.y = p[1];
  return a;
}
// B (4x16, KxN, row-major src): lanes 0-15 -> N=lane, VGPR0/1 = K0/K1;
//                               lanes 16-31 -> N=lane-16, VGPR0/1 = K2/K3.
__device__ __forceinline__ v2f ldB(const float* S, int ld, int k0, int c0, int lane) {
  int n  = lane & 15;
  int kk = k0 + ((lane >> 4) << 1);
  v2f b;
  b.x = S[(size_t)kk       * ld + c0 + n];
  b.y = S[(size_t)(kk + 1) * ld + c0 + n];
  return b;
}
// C/D (16x16 f32): VGPR r: lanes 0-15 -> (M=r, N=lane); lanes 16-31 -> (M=8+r, N=lane-16).

__device__ __forceinline__ v8f wmma_f32(v2f a, v2f b, v8f c) {
  return __builtin_amdgcn_wmma_f32_16x16x4_f32(false, a, false, b, (short)0, c, false, false);
}

__device__ __forceinline__ float elu1(float t) {
  return t > 0.f ? t : (__expf(t) - 1.f);
}

// ---- K1: projections --------------------------------------------------------
// 2048 waves: which(2) x rowT(128) x colT(8). Each wave: one 16x16 tile, K=128.
__global__ __launch_bounds__(128) void k_proj(const float* __restrict__ H,
                                              const float* __restrict__ dW1,
                                              const float* __restrict__ db1,
                                              float* __restrict__ hi,
                                              float* __restrict__ hjT) {
  int waveId = blockIdx.x * (blockDim.x >> 5) + (threadIdx.x >> 5);
  int lane = threadIdx.x & 31;
  int colT  = waveId & 7;
  int rowT  = (waveId >> 3) & 127;
  int which = waveId >> 10;               // 0 -> Hi, 1 -> Hj
  int r0 = rowT * 16;                     // row over flattened [B*N]
  int c0 = colT * 16;                     // f column
  const float* W = dW1 + (size_t)which * FF * FF;  // dW1 [256][128] row-major

  v8f c = {};
  for (int k = 0; k < FF; k += 4) {
    v2f a = ldA(H, FF, r0, k, lane);
    v2f b = ldB(W, FF, k, c0, lane);
    c = wmma_f32(a, b, c);
  }

  int col   = c0 + (lane & 15);
  int mbase = r0 + ((lane >> 4) << 3);
#pragma unroll
  for (int r = 0; r < 8; ++r) {
    int mrow = mbase + r;                 // flattened b*N + n
    if (which == 0) {
      hi[(size_t)mrow * FF + col] = c[r] + db1[col];   // fold db1 into Hi
    } else {
      int b = mrow >> 9;                  // / N
      int n = mrow & (NN - 1);
      hjT[((size_t)b * FF + col) * NN + n] = c[r];     // f-major transpose
    }
  }
}

// ---- K2: pairwise contradiction scores (the hot kernel) ---------------------
// Grid (4, 64), block 512: thread t <-> column j; each WG handles 8 rows i.
// Per f: one coalesced HjT load, 8 uniform (scalar) Hi loads, 8x (add+elu+fma).
#define IBLK 8
__global__ __launch_bounds__(512) void k_pair(const float* __restrict__ hi,
                                              const float* __restrict__ hjT,
                                              const float* __restrict__ A,
                                              const float* __restrict__ dW2,
                                              const float* __restrict__ db2,
                                              float* __restrict__ maxc,
                                              float* __restrict__ deg,
                                              float* __restrict__ outMask) {
  int b  = blockIdx.x;
  int i0 = blockIdx.y * IBLK;
  int j  = threadIdx.x;
  const float* hjb = hjT + (size_t)b * FF * NN;
  const float* hib = hi  + ((size_t)b * NN + i0) * FF;

  float acc[IBLK];
#pragma unroll
  for (int ii = 0; ii < IBLK; ++ii) acc[ii] = 0.f;

#pragma unroll 2
  for (int f = 0; f < FF; ++f) {
    float hj = hjb[(size_t)f * NN + j];   // coalesced, L2-resident
    float w  = dW2[f];                    // uniform -> scalar load
#pragma unroll
    for (int ii = 0; ii < IBLK; ++ii) {
      float t = hj + hib[(size_t)ii * FF + f];   // uniform -> scalar load
      acc[ii] = fmaf(elu1(t), w, acc[ii]);
    }
  }

  __shared__ float redMax[16][IBLK];
  __shared__ float redSum[16][IBLK];
  int wave = threadIdx.x >> 5;
  int lane = threadIdx.x & 31;
  float bias2 = db2[0];

#pragma unroll
  for (int ii = 0; ii < IBLK; ++ii) {
    float a = A[((size_t)b * NN + i0 + ii) * NN + j];
    float s = 1.f / (1.f + __expf(-(acc[ii] + bias2)));   // sigmoid
    float contra = (a > 0.1f) ? s : 0.f;                  // CONNECT_EPS
    float nb     = (a > 0.f) ? 1.f : 0.f;                 // strict >0 for degree
#pragma unroll
    for (int off = 16; off; off >>= 1) {
      contra = fmaxf(contra, __shfl_xor(contra, off, 32));
      nb += __shfl_xor(nb, off, 32);
    }
    if (lane == 0) { redMax[wave][ii] = contra; redSum[wave][ii] = nb; }
  }
  __syncthreads();

  if (threadIdx.x < IBLK) {
    int ii = threadIdx.x;
    float m = redMax[0][ii];
    float s = redSum[0][ii];
#pragma unroll
    for (int w = 1; w < 16; ++w) {
      m = fmaxf(m, redMax[w][ii]);
      s += redSum[w][ii];
    }
    int gi = b * NN + i0 + ii;
    maxc[gi] = m;
    deg[gi]  = s;
    outMask[gi] = (m > 0.5f) ? 1.f : 0.f;   // THRESHOLD
  }
}

// ---- K3: neighbor mean  nbrF = ((A>0) @ H) / max(deg,1) ---------------------
// 1024 waves: b(4) x rowT(32) x colT(8).
__global__ __launch_bounds__(128) void k_nbr(const float* __restrict__ A,
                                             const float* __restrict__ H,
                                             const float* __restrict__ deg,
                                             float* __restrict__ nbrF) {
  int waveId = blockIdx.x * (blockDim.x >> 5) + (threadIdx.x >> 5);
  int lane = threadIdx.x & 31;
  int colT = waveId & 7;
  int rowT = (waveId >> 3) & 31;
  int b    = waveId >> 8;
  const float* Ab = A + (size_t)b * NN * NN;
  const float* Hb = H + (size_t)b * NN * FF;
  int r0 = rowT * 16, c0 = colT * 16;

  int m     = lane & 15;
  int khalf = (lane >> 4) << 1;
  v8f c = {};
  for (int k = 0; k < NN; k += 4) {
    const float* pa = Ab + (size_t)(r0 + m) * NN + k + khalf;
    v2f a;
    a.x = (pa[0] > 0.f) ? 1.f : 0.f;     // binarize adjacency on the fly
    a.y = (pa[1] > 0.f) ? 1.f : 0.f;
    v2f bf = ldB(Hb, FF, k, c0, lane);
    c = wmma_f32(a, bf, c);
  }

  int col   = c0 + (lane & 15);
  int mbase = r0 + ((lane >> 4) << 3);
#pragma unroll
  for (int r = 0; r < 8; ++r) {
    int n = mbase + r;
    float d = deg[b * NN + n];
    nbrF[((size_t)b * NN + n) * FF + col] = c[r] / fmaxf(d, 1.f);
  }
}

// ---- K4a: res1 = elu(H*rW1a + nbrF*rW1b + maxc*rW1[256] + rb1) --------------
// 1024 waves: rowT(128) x colT(8) over flattened [2048 x 128].
__global__ __launch_bounds__(128) void k_res1(const float* __restrict__ H,
                                              const float* __restrict__ nbrF,
                                              const float* __restrict__ maxc,
                                              const float* __restrict__ rW1,
                                              const float* __restrict__ rb1,
                                              float* __restrict__ res1) {
  int waveId = blockIdx.x * (blockDim.x >> 5) + (threadIdx.x >> 5);
  int lane = threadIdx.x & 31;
  int colT = waveId & 7;
  int rowT = (waveId >> 3) & 127;
  int r0 = rowT * 16, c0 = colT * 16;

  v8f c = {};
  for (int k = 0; k < FF; k += 4) {              // H part: rW1 rows [0,128)
    v2f a = ldA(H, FF, r0, k, lane);
    v2f b = ldB(rW1, FF, k, c0, lane);
    c = wmma_f32(a, b, c);
  }
  for (int k = 0; k < FF; k += 4) {              // nbrF part: rW1 rows [128,256)
    v2f a = ldA(nbrF, FF, r0, k, lane);
    v2f b = ldB(rW1 + (size_t)FF * FF, FF, k, c0, lane);
    c = wmma_f32(a, b, c);
  }

  int col   = c0 + (lane & 15);
  int mbase = r0 + ((lane >> 4) << 3);
  float wlast = rW1[(size_t)256 * FF + col];     // rank-1 max_contra column
  float bias  = rb1[col];
#pragma unroll
  for (int r = 0; r < 8; ++r) {
    int g = mbase + r;                           // flattened b*N + n
    float v = c[r] + maxc[g] * wlast + bias;
    res1[(size_t)g * FF + col] = elu1(v);
  }
}

// ---- K4b: res2 = res1*rW2 + rb2; out = update ? res2 : H --------------------
__global__ __launch_bounds__(128) void k_res2(const float* __restrict__ res1,
                                              const float* __restrict__ rW2,
                                              const float* __restrict__ rb2,
                                              const float* __restrict__ maxc,
                                              const float* __restrict__ deg,
                                              const float* __restrict__ H,
                                              float* __restrict__ out) {
  int waveId = blockIdx.x * (blockDim.x >> 5) + (threadIdx.x >> 5);
  int lane = threadIdx.x & 31;
  int colT = waveId & 7;
  int rowT = (waveId >> 3) & 127;
  int r0 = rowT * 16, c0 = colT * 16;

  v8f c = {};
  for (int k = 0; k < FF; k += 4) {
    v2f a = ldA(res1, FF, r0, k, lane);
    v2f b = ldB(rW2, FF, k, c0, lane);
    c = wmma_f32(a, b, c);
  }

  int col   = c0 + (lane & 15);
  int mbase = r0 + ((lane >> 4) << 3);
  float bias = rb2[col];
#pragma unroll
  for (int r = 0; r < 8; ++r) {
    int g = mbase + r;
    bool upd = (maxc[g] > 0.5f) && (deg[g] > 0.f);
    float v = c[r] + bias;
    out[(size_t)g * FF + col] = upd ? v : H[(size_t)g * FF + col];
  }
}

// ---- launcher ---------------------------------------------------------------
extern "C" void kernel_launch(void* const* d_in, const int* in_sizes, int n_in,
                              void* d_out, int out_size, void* d_ws, size_t ws_size,
                              hipStream_t stream) {
  const float* H   = (const float*)d_in[0];
  const float* A   = (const float*)d_in[1];
  const float* dW1 = (const float*)d_in[2];
  const float* db1 = (const float*)d_in[3];
  const float* dW2 = (const float*)d_in[4];
  const float* db2 = (const float*)d_in[5];
  const float* rW1 = (const float*)d_in[6];
  const float* rb1 = (const float*)d_in[7];
  const float* rW2 = (const float*)d_in[8];
  const float* rb2 = (const float*)d_in[9];

  float* out = (float*)d_out;                 // [262144] H_resolved + [2048] mask
  float* ws  = (float*)d_ws;
  float* hi   = ws;                           // [B*N*F] = 262144
  float* hjT  = ws + 262144;                  // [B*F*N] = 262144
  float* nbrF = ws + 524288;                  // 262144
  float* res1 = ws + 786432;                  // 262144
  float* maxc = ws + 1048576;                 // 2048
  float* deg  = ws + 1050624;                 // 2048

  // K1: 2048 waves (Hi + HjT tiles)
  k_proj<<<512, 128, 0, stream>>>(H, dW1, db1, hi, hjT);

  // K2: hot pairwise kernel; grid (B, N/IBLK) x 512 threads
  dim3 g2(BB, NN / IBLK);
  k_pair<<<g2, 512, 0, stream>>>(hi, hjT, A, dW2, db2, maxc, deg, out + 262144);

  // K3: neighbor-mean GEMM, 1024 waves
  k_nbr<<<256, 128, 0, stream>>>(A, H, deg, nbrF);

  // K4: resolver MLP, 1024 waves each
  k_res1<<<256, 128, 0, stream>>>(H, nbrF, maxc, rW1, rb1, res1);
  k_res2<<<256, 128, 0, stream>>>(res1, rW2, rb2, maxc, deg, H, out);
}